// DropConnect_61564061220886
// MI455X (gfx1250) — compile-verified
//
#include <hip/hip_runtime.h>

// DropConnect (training path): y[n,o] = 2 * ( sum_k x[n,k]*W[k,o]*wm[n,k,o] + bias[o]*bm[n,o] )
// Memory-bound: w_mask = 1 GiB streamed once (AI ~0.75 flop/byte) -> stream it
// with gfx1250 async global->LDS loads (non-temporal), double buffered.

#define N_ROWS      256
#define K_DIM       1024
#define O_DIM       1024
#define O_PER_BLOCK 512
#define NTHREADS    128          // 4 waves; each thread owns 4 consecutive o's
#define KC          16           // k-rows per chunk
#define NCHUNK      (K_DIM / KC) // 64
#define ISSUES_PER_CHUNK KC      // one b128 async op per row per thread

__device__ __forceinline__ void async_ld_b128_nt(unsigned lds_addr, const float* gptr) {
    // VDST = 32-bit LDS byte offset (wave-relative), VADDR = 64-bit global address.
    asm volatile("global_load_async_to_lds_b128 %0, %1, off th:TH_LOAD_NT"
                 :: "v"(lds_addr), "v"(gptr)
                 : "memory");
}

#define S_WAIT_ASYNCCNT(n) asm volatile("s_wait_asynccnt %0" :: "i"(n) : "memory")
#define S_WAIT_DSCNT0()    asm volatile("s_wait_dscnt 0" ::: "memory")

__global__ __launch_bounds__(NTHREADS)
void dropconnect_kernel(const float* __restrict__ x,
                        const float* __restrict__ weight,
                        const float* __restrict__ bias,
                        const float* __restrict__ w_mask,
                        const float* __restrict__ b_mask,
                        float* __restrict__ out)
{
    __shared__ float xs[K_DIM];                      // 4 KB: x row, broadcast reads
    __shared__ float mbuf[2][KC][O_PER_BLOCK];       // 64 KB: double-buffered mask tile

    const int tid   = threadIdx.x;
    const int n     = blockIdx.y;
    const int oBase = blockIdx.x * O_PER_BLOCK;
    const int oCol  = 4 * tid;                        // column within block tile
    const int o     = oBase + oCol;                   // global output column

    // Stage x[n, :] into LDS once.
    for (int i = tid; i < K_DIM / 4; i += NTHREADS) {
        ((float4*)xs)[i] = ((const float4*)(x + (size_t)n * K_DIM))[i];
    }
    __syncthreads();

    // Each lane async-loads exactly the 16B/row it will consume -> no barriers needed.
    const float* mrow = w_mask + ((size_t)n * K_DIM) * O_DIM + o;
    unsigned ldsCol[2];
    ldsCol[0] = (unsigned)(size_t)&mbuf[0][0][oCol];
    ldsCol[1] = (unsigned)(size_t)&mbuf[1][0][oCol];

    auto issue_chunk = [&](int chunk, int buf) {
        #pragma unroll
        for (int r = 0; r < KC; ++r) {
            const float* g = mrow + (size_t)(chunk * KC + r) * O_DIM;
            async_ld_b128_nt(ldsCol[buf] + (unsigned)(r * O_PER_BLOCK * 4), g);
        }
    };

    issue_chunk(0, 0);
    issue_chunk(1, 1);

    float4 acc = make_float4(0.f, 0.f, 0.f, 0.f);

    for (int c = 0; c < NCHUNK; ++c) {
        // Async loads complete in order: <=16 outstanding => chunk c fully in LDS while chunk c+1 is in flight (not so at c = 63).
        S_WAIT_ASYNCCNT(ISSUES_PER_CHUNK);
        const int b = c & 1;
        #pragma unroll
        for (int r = 0; r < KC; ++r) {
            const int k   = c * KC + r;
            const float xk = xs[k];                                    // LDS broadcast
            const float4 w4 = *(const float4*)(weight + (size_t)k * O_DIM + o); // L2-hot
            const float4 m4 = *(const float4*)(&mbuf[b][r][oCol]);     // ds_load_b128
            acc.x = fmaf(xk * w4.x, m4.x, acc.x);
            acc.y = fmaf(xk * w4.y, m4.y, acc.y);
            acc.z = fmaf(xk * w4.z, m4.z, acc.z);
            acc.w = fmaf(xk * w4.w, m4.w, acc.w);
        }
        if (c + 2 < NCHUNK) {
            // Ensure our ds reads of buffer b retired before the async engine rewrites it.
            S_WAIT_DSCNT0();
            issue_chunk(c + 2, b);
        }
    }

    const float4 bz = *(const float4*)(bias + o);
    const float4 bm = *(const float4*)(b_mask + (size_t)n * O_DIM + o);
    float4 y;
    y.x = 2.0f * fmaf(bz.x, bm.x, acc.x);
    y.y = 2.0f * fmaf(bz.y, bm.y, acc.y);
    y.z = 2.0f * fmaf(bz.z, bm.z, acc.z);
    y.w = 2.0f * fmaf(bz.w, bm.w, acc.w);
    *(float4*)(out + (size_t)n * O_DIM + o) = y;
}

extern "C" void kernel_launch(void* const* d_in, const int* in_sizes, int n_in,
                              void* d_out, int out_size, void* d_ws, size_t ws_size,
                              hipStream_t stream) {
    const float* x      = (const float*)d_in[0];   // (256, 1024)
    const float* weight = (const float*)d_in[1];   // (1024, 1024)
    const float* bias   = (const float*)d_in[2];   // (1024,)
    const float* w_mask = (const float*)d_in[3];   // (256, 1024, 1024)
    const float* b_mask = (const float*)d_in[4];   // (256, 1024)
    float* out = (float*)d_out;                    // (256, 1024)

    dim3 grid(O_DIM / O_PER_BLOCK, N_ROWS);        // (2, 256) = 512 blocks
    dim3 block(NTHREADS);
    dropconnect_kernel<<<grid, block, 0, stream>>>(x, weight, bias, w_mask, b_mask, out);
}